// AngleFreqEnhance_74019466379690
// MI455X (gfx1250) — compile-verified
//
#include <hip/hip_runtime.h>
#include <math.h>

#define B_   16
#define C_   256
#define M_   16
#define H_   128
#define W_   128
#define HW_  (H_ * W_)
#define NBINS 180
#define PI_F 3.14159265358979f
#define BW_F 0.26179938779914946f   /* radians(15 deg) */
#define RHI  19.2f                  /* 0.3 * r_max, r_max = 64 */
#define INV_N2 (1.0f / 16384.0f)    /* combined fwd+inv ortho scale, folded into gain */

typedef __attribute__((ext_vector_type(2))) float v2f;
typedef __attribute__((ext_vector_type(4))) float v4f;
typedef __attribute__((ext_vector_type(8))) float v8f;

__device__ __forceinline__ int bitrev7(int v) {
  return (int)(__brev((unsigned)v) >> 25);
}

// ---------------------------------------------------------------------------
// K1: x_proj[b,m,hw] = sum_c w_in[m,c] * x[b,c,hw]
// One wave computes a 16(m) x 16(n) tile via 64 chained V_WMMA_F32_16X16X4_F32.
// A (16x4 f32): lane<16 holds M=lane,K={0,1}; lane>=16 holds M=lane-16,K={2,3}
// B (4x16 f32): lane<16 holds K={0,1},N=lane; lane>=16 holds K={2,3},N=lane-16
// D (16x16 f32): VGPR r -> M = r + 8*(lane>=16), N = lane&15
// ---------------------------------------------------------------------------
__global__ void __launch_bounds__(128) k_proj_in(const float* __restrict__ x,
                                                 const float* __restrict__ w_in,
                                                 float* __restrict__ xproj) {
  __shared__ float lw[M_ * 257];                 // padded: stride 257 -> no bank conflicts
  for (int i = threadIdx.x; i < M_ * C_; i += 128) {
    int m = i >> 8, c = i & 255;
    lw[m * 257 + c] = w_in[i];
  }
  __syncthreads();

  const int lane  = threadIdx.x & 31;
  const int wave  = threadIdx.x >> 5;
  const int tile  = blockIdx.x * 4 + wave;       // 16384 tiles total
  const int b     = tile >> 10;                  // 1024 n-tiles per batch
  const int n0    = (tile & 1023) << 4;
  const int lrow  = lane & 15;
  const int khalf = (lane >> 4) << 1;            // 0 or 2

  const float* xb = x + (size_t)b * C_ * HW_ + n0 + lrow;
  v8f acc = {};
  #pragma unroll 4
  for (int k = 0; k < C_; k += 4) {
    v2f a, bf;
    a.x  = lw[lrow * 257 + k + khalf];
    a.y  = lw[lrow * 257 + k + khalf + 1];
    bf.x = xb[(size_t)(k + khalf) * HW_];
    bf.y = xb[(size_t)(k + khalf + 1) * HW_];
    acc = __builtin_amdgcn_wmma_f32_16x16x4_f32(false, a, false, bf,
                                                (short)0, acc, false, false);
  }
  float* o = xproj + (size_t)b * M_ * HW_ + n0 + lrow;
  const int mbase = (lane >> 4) << 3;
  #pragma unroll
  for (int r = 0; r < 8; ++r) o[(size_t)(mbase + r) * HW_] = acc[r];
}

// ---------------------------------------------------------------------------
// K2: forward 2D FFT, one plane (b,m) per workgroup (unscaled; ortho factor
// folded into the inverse-side gain). Row-DIF then column-DIF; output left
// bit-reversed in BOTH dims (no reorder pass).
// Plane staged in LDS stride-128 (row pass: lanes hit distinct banks every
// stage; column pass: lanes sweep consecutive columns) -> conflict-free, and
// rows are 16B aligned for B128 async transfers.
// Global->LDS via GLOBAL_LOAD_ASYNC_TO_LDS_B128, LDS->global via
// GLOBAL_STORE_ASYNC_FROM_LDS_B128 (ASYNCcnt; S_ENDPGM implicitly waits).
// ---------------------------------------------------------------------------
__global__ void __launch_bounds__(256) k_fft_fwd(const float* __restrict__ xproj,
                                                 float* __restrict__ fre,
                                                 float* __restrict__ fim) {
  extern __shared__ float smem[];                // dynamic LDS starts at offset 0
  float* re = smem;                              // bytes [0, 65536)
  float* im = smem + HW_;                        // bytes [65536, 131072)
  const int plane = blockIdx.x;
  const float* src = xproj + (size_t)plane * HW_;

  // zero imaginary part (ds stores overlap with the async loads below)
  #pragma unroll
  for (int i = 0; i < HW_ / (256 * 4); ++i) {
    v4f z = {0.0f, 0.0f, 0.0f, 0.0f};
    *(v4f*)(im + (i * 256 + threadIdx.x) * 4) = z;
  }
  // async copy real part: global -> LDS, 16B per lane per op
  #pragma unroll
  for (int i = 0; i < HW_ / (256 * 4); ++i) {
    unsigned off = (unsigned)((i * 256 + threadIdx.x) * 16);  // byte offset
    asm volatile("global_load_async_to_lds_b128 %0, %1, %2"
                 :: "v"(off), "v"(off), "s"(src) : "memory");
  }
  asm volatile("s_wait_asynccnt 0x0" ::: "memory");
  __syncthreads();

  // row pass (transform along c)
  for (int sh = 6; sh >= 0; --sh) {
    const int half = 1 << sh;
    for (int q = threadIdx.x; q < HW_ / 2; q += 256) {
      int row = q >> 6, t = q & 63;
      int pos = t & (half - 1);
      int i0  = ((t >> sh) << (sh + 1)) + pos;
      int a0 = (row << 7) + i0, a1 = a0 + half;
      float ur = re[a0], ui = im[a0], vr = re[a1], vi = im[a1];
      float s, cc;
      __sincosf(-PI_F * (float)pos / (float)half, &s, &cc);
      float dr = ur - vr, di = ui - vi;
      re[a0] = ur + vr;            im[a0] = ui + vi;
      re[a1] = dr * cc - di * s;   im[a1] = dr * s + di * cc;
    }
    __syncthreads();
  }
  // column pass (transform along r)
  for (int sh = 6; sh >= 0; --sh) {
    const int half = 1 << sh;
    for (int q = threadIdx.x; q < HW_ / 2; q += 256) {
      int col = q & 127, t = q >> 7;
      int pos = t & (half - 1);
      int i0  = ((t >> sh) << (sh + 1)) + pos;
      int a0 = (i0 << 7) + col, a1 = ((i0 + half) << 7) + col;
      float ur = re[a0], ui = im[a0], vr = re[a1], vi = im[a1];
      float s, cc;
      __sincosf(-PI_F * (float)pos / (float)half, &s, &cc);
      float dr = ur - vr, di = ui - vi;
      re[a0] = ur + vr;            im[a0] = ui + vi;
      re[a1] = dr * cc - di * s;   im[a1] = dr * s + di * cc;
    }
    __syncthreads();
  }

  // async copy out: LDS -> global (spectrum, bit-reversed storage order)
  float* dre  = fre + (size_t)plane * HW_;
  float* dim_ = fim + (size_t)plane * HW_;
  #pragma unroll
  for (int i = 0; i < HW_ / (256 * 4); ++i) {
    unsigned off  = (unsigned)((i * 256 + threadIdx.x) * 16);
    unsigned ldsi = off + (unsigned)(HW_ * 4);
    asm volatile("global_store_async_from_lds_b128 %0, %1, %2"
                 :: "v"(off), "v"(off), "s"(dre) : "memory");
    asm volatile("global_store_async_from_lds_b128 %0, %1, %2"
                 :: "v"(off), "v"(ldsi), "s"(dim_) : "memory");
  }
  // S_ENDPGM performs an implicit S_WAIT_IDLE (drains ASYNCcnt)
}

// ---------------------------------------------------------------------------
// K3: zero the energy histogram
// ---------------------------------------------------------------------------
__global__ void k_zero(float* __restrict__ p, int n) {
  int i = blockIdx.x * blockDim.x + threadIdx.x;
  if (i < n) p[i] = 0.0f;
}

// ---------------------------------------------------------------------------
// K4: angular energy histogram of high-freq magnitude (avg over 16 channels).
// Storage index (r,c) is bit-reversed; natural shifted coords recovered via
// brev7 + (+64 mod 128) -- fftshift folded into index math. The uniform 128x
// magnitude scale (unscaled forward FFT) cancels in all peak decisions.
// ---------------------------------------------------------------------------
__global__ void __launch_bounds__(256) k_hist(const float* __restrict__ fre,
                                              const float* __restrict__ fim,
                                              float* __restrict__ energy) {
  __shared__ float h[NBINS];
  for (int i = threadIdx.x; i < NBINS; i += 256) h[i] = 0.0f;
  __syncthreads();

  const int b   = blockIdx.y;
  const int idx = blockIdx.x * 256 + threadIdx.x;   // grid.x = 64 -> 16384
  const int r = idx >> 7, c = idx & 127;
  const int hs  = (bitrev7(r) + 64) & 127;
  const int wsh = (bitrev7(c) + 64) & 127;
  const float dy = (float)(hs - 64), dx = (float)(wsh - 64);
  const float rad = sqrtf(dy * dy + dx * dx);
  if (rad > RHI) {
    const float* pr = fre + (size_t)b * M_ * HW_ + idx;
    const float* pi = fim + (size_t)b * M_ * HW_ + idx;
    float msum = 0.0f;
    #pragma unroll
    for (int m = 0; m < M_; ++m) {
      float a = pr[(size_t)m * HW_], bb = pi[(size_t)m * HW_];
      msum += sqrtf(a * a + bb * bb);
    }
    msum *= (1.0f / 16.0f);
    float theta = atan2f(dy, dx) + PI_F;            // [0, 2pi]
    float tm = fmodf(theta, PI_F);
    int bin = (int)floorf(tm * ((float)NBINS / PI_F));
    bin = bin < 0 ? 0 : (bin > NBINS - 1 ? NBINS - 1 : bin);
    atomicAdd(&h[bin], msum);                       // ds_add_f32
  }
  __syncthreads();
  for (int i = threadIdx.x; i < NBINS; i += 256)
    atomicAdd(&energy[b * NBINS + i], h[i]);        // global_atomic_add_f32
}

// ---------------------------------------------------------------------------
// K5: smooth histogram, pick <=2 circular local peaks (> mean), build gain map
// gain = alpha in peak wedges (high annulus), beta elsewhere in annulus, 1 inside.
// ---------------------------------------------------------------------------
__global__ void __launch_bounds__(256) k_peaks_gain(const float* __restrict__ energy,
                                                    float* __restrict__ gain) {
  const int b = blockIdx.x;
  __shared__ float e[NBINS + 2];
  __shared__ float es[NBINS];
  __shared__ float pa[2];
  if (threadIdx.x < NBINS) e[threadIdx.x + 1] = energy[b * NBINS + threadIdx.x];
  if (threadIdx.x == 0) { e[0] = 0.0f; e[NBINS + 1] = 0.0f; }  // zero padding
  __syncthreads();
  if (threadIdx.x < NBINS)
    es[threadIdx.x] = 0.25f * e[threadIdx.x] + 0.5f * e[threadIdx.x + 1]
                    + 0.25f * e[threadIdx.x + 2];
  __syncthreads();
  if (threadIdx.x == 0) {
    float mean = 0.0f;
    for (int i = 0; i < NBINS; ++i) mean += es[i];
    mean *= (1.0f / (float)NBINS);
    float b1 = -INFINITY, b2 = -INFINITY;
    int i1 = -1, i2 = -1, amax = 0;
    float av = es[0];
    for (int i = 0; i < NBINS; ++i) {
      if (es[i] > av) { av = es[i]; amax = i; }
      float pv = es[(i + NBINS - 1) % NBINS];     // circular (jnp.roll)
      float nx = es[(i + 1) % NBINS];
      if (es[i] > mean && es[i] > pv && es[i] > nx) {
        if (es[i] > b1) { b2 = b1; i2 = i1; b1 = es[i]; i1 = i; }
        else if (es[i] > b2) { b2 = es[i]; i2 = i; }
      }
    }
    int f0, f1;
    if (i1 < 0)      { f0 = amax; f1 = amax; }    // fallback argmax
    else if (i2 < 0) { f0 = i1;   f1 = i1;   }    // pad with last valid peak
    else             { f0 = i1;   f1 = i2;   }
    pa[0] = ((float)f0 + 0.5f) * (PI_F / (float)NBINS);
    pa[1] = ((float)f1 + 0.5f) * (PI_F / (float)NBINS);
  }
  __syncthreads();
  const float p0 = pa[0], p1 = pa[1];
  float* g = gain + (size_t)b * HW_;
  for (int idx = threadIdx.x; idx < HW_; idx += 256) {
    int y = idx >> 7, xx = idx & 127;
    float dy = (float)(y - 64), dx = (float)(xx - 64);
    float rad = sqrtf(dy * dy + dx * dx);
    bool high = rad > RHI;
    float theta = atan2f(dy, dx) + PI_F;
    float d0 = fabsf(theta - p0); d0 = fminf(d0, PI_F - d0);
    float d1 = fabsf(theta - p1); d1 = fminf(d1, PI_F - d1);
    bool enh = high && (d0 <= BW_F || d1 <= BW_F);
    g[idx] = enh ? 1.2f : (high ? 0.8f : 1.0f);
  }
}

// ---------------------------------------------------------------------------
// K6: apply (gain * 1/N^2) while loading (bit-reversal + fftshift folded into
// the gain lookup), inverse 2D FFT (col-DIT then row-DIT; bit-reversed input
// -> natural output), async-store the real part LDS -> global.
// ---------------------------------------------------------------------------
__global__ void __launch_bounds__(256) k_fft_inv(const float* __restrict__ fre,
                                                 const float* __restrict__ fim,
                                                 const float* __restrict__ gain,
                                                 float* __restrict__ xenh) {
  extern __shared__ float smem[];
  float* re = smem;
  float* im = smem + HW_;
  const int plane = blockIdx.x;
  const int b = plane >> 4;
  const float* sre = fre + (size_t)plane * HW_;
  const float* sim = fim + (size_t)plane * HW_;
  const float* g = gain + (size_t)b * HW_;

  for (int idx = threadIdx.x; idx < HW_; idx += 256) {
    int r = idx >> 7, c = idx & 127;
    int hs  = (bitrev7(r) + 64) & 127;
    int wsh = (bitrev7(c) + 64) & 127;
    float gg = g[hs * 128 + wsh] * INV_N2;       // gain + both ortho scales
    re[idx] = sre[idx] * gg;
    im[idx] = sim[idx] * gg;
  }
  __syncthreads();

  // column DIT
  for (int sh = 0; sh <= 6; ++sh) {
    const int half = 1 << sh;
    for (int q = threadIdx.x; q < HW_ / 2; q += 256) {
      int col = q & 127, t = q >> 7;
      int pos = t & (half - 1);
      int i0  = ((t >> sh) << (sh + 1)) + pos;
      int a0 = (i0 << 7) + col, a1 = ((i0 + half) << 7) + col;
      float ur = re[a0], ui = im[a0], vr = re[a1], vi = im[a1];
      float s, cc;
      __sincosf(PI_F * (float)pos / (float)half, &s, &cc);
      float tr = vr * cc - vi * s, ti = vr * s + vi * cc;
      re[a0] = ur + tr; im[a0] = ui + ti;
      re[a1] = ur - tr; im[a1] = ui - ti;
    }
    __syncthreads();
  }
  // row DIT
  for (int sh = 0; sh <= 6; ++sh) {
    const int half = 1 << sh;
    for (int q = threadIdx.x; q < HW_ / 2; q += 256) {
      int row = q >> 6, t = q & 63;
      int pos = t & (half - 1);
      int i0  = ((t >> sh) << (sh + 1)) + pos;
      int a0 = (row << 7) + i0, a1 = a0 + half;
      float ur = re[a0], ui = im[a0], vr = re[a1], vi = im[a1];
      float s, cc;
      __sincosf(PI_F * (float)pos / (float)half, &s, &cc);
      float tr = vr * cc - vi * s, ti = vr * s + vi * cc;
      re[a0] = ur + tr; im[a0] = ui + ti;
      re[a1] = ur - tr; im[a1] = ui - ti;
    }
    __syncthreads();
  }

  // async copy real part LDS -> global (imag discarded, matches .real)
  float* o = xenh + (size_t)plane * HW_;
  #pragma unroll
  for (int i = 0; i < HW_ / (256 * 4); ++i) {
    unsigned off = (unsigned)((i * 256 + threadIdx.x) * 16);
    asm volatile("global_store_async_from_lds_b128 %0, %1, %2"
                 :: "v"(off), "v"(off), "s"(o) : "memory");
  }
}

// ---------------------------------------------------------------------------
// K7: out[b,c,hw] = x[b,c,hw] + sum_m w_out[c,m] * xenh[b,m,hw]
// One wave per 16-wide spatial tile: load B fragments once (K=16 -> 4 steps),
// sweep all 16 channel tiles (64 WMMAs), residual-add on store.
// ---------------------------------------------------------------------------
__global__ void __launch_bounds__(128) k_proj_out(const float* __restrict__ x,
                                                  const float* __restrict__ w_out,
                                                  const float* __restrict__ xenh,
                                                  float* __restrict__ out) {
  __shared__ float lw[C_ * 17];                  // padded: stride 17 -> no conflicts
  for (int i = threadIdx.x; i < C_ * M_; i += 128) {
    int c = i >> 4, m = i & 15;
    lw[c * 17 + m] = w_out[i];
  }
  __syncthreads();

  const int lane  = threadIdx.x & 31;
  const int wave  = threadIdx.x >> 5;
  const int tile  = blockIdx.x * 4 + wave;       // 16384 tiles
  const int b     = tile >> 10;
  const int n0    = (tile & 1023) << 4;
  const int lrow  = lane & 15;
  const int khalf = (lane >> 4) << 1;

  const float* src = xenh + (size_t)b * M_ * HW_ + n0 + lrow;
  v2f bfrag[4];
  #pragma unroll
  for (int ks = 0; ks < 4; ++ks) {
    bfrag[ks].x = src[(size_t)(4 * ks + khalf) * HW_];
    bfrag[ks].y = src[(size_t)(4 * ks + khalf + 1) * HW_];
  }

  const int mbase = (lane >> 4) << 3;
  const float* xr = x + (size_t)b * C_ * HW_ + n0 + lrow;
  float* orow = out + (size_t)b * C_ * HW_ + n0 + lrow;

  #pragma unroll
  for (int ct = 0; ct < 16; ++ct) {
    const int c0 = ct << 4;
    v8f acc = {};
    #pragma unroll
    for (int ks = 0; ks < 4; ++ks) {
      v2f a;
      a.x = lw[(c0 + lrow) * 17 + 4 * ks + khalf];
      a.y = lw[(c0 + lrow) * 17 + 4 * ks + khalf + 1];
      acc = __builtin_amdgcn_wmma_f32_16x16x4_f32(false, a, false, bfrag[ks],
                                                  (short)0, acc, false, false);
    }
    #pragma unroll
    for (int r = 0; r < 8; ++r) {
      size_t off = (size_t)(c0 + mbase + r) * HW_;
      orow[off] = xr[off] + acc[r];              // residual add
    }
  }
}

// ---------------------------------------------------------------------------
extern "C" void kernel_launch(void* const* d_in, const int* in_sizes, int n_in,
                              void* d_out, int out_size, void* d_ws, size_t ws_size,
                              hipStream_t stream) {
  (void)in_sizes; (void)n_in; (void)out_size; (void)ws_size;
  const float* x     = (const float*)d_in[0];   // [16,256,128,128]
  const float* w_in  = (const float*)d_in[1];   // [16,256]
  const float* w_out = (const float*)d_in[2];   // [256,16]
  float* out = (float*)d_out;                   // [16,256,128,128]
  float* ws  = (float*)d_ws;

  const size_t PLANES = (size_t)B_ * M_ * HW_;  // 4,194,304 floats
  float* xproj  = ws;                           // reused as x_enh after K6
  float* fre    = ws + PLANES;
  float* fim    = ws + 2 * PLANES;
  float* gain   = ws + 3 * PLANES;              // B*HW floats
  float* energy = gain + (size_t)B_ * HW_;      // B*180 floats

  const size_t fftLds = (size_t)(2 * HW_) * sizeof(float);  // 131,072 B

  k_zero      <<<(B_ * NBINS + 255) / 256, 256, 0, stream>>>(energy, B_ * NBINS);
  k_proj_in   <<<4096, 128, 0, stream>>>(x, w_in, xproj);
  k_fft_fwd   <<<B_ * M_, 256, fftLds, stream>>>(xproj, fre, fim);
  k_hist      <<<dim3(64, 16), 256, 0, stream>>>(fre, fim, energy);
  k_peaks_gain<<<B_, 256, 0, stream>>>(energy, gain);
  k_fft_inv   <<<B_ * M_, 256, fftLds, stream>>>(fre, fim, gain, xproj);
  k_proj_out  <<<4096, 128, 0, stream>>>(x, w_out, xproj, out);
}